// Net_5841155522745
// MI455X (gfx1250) — compile-verified
//
#include <hip/hip_runtime.h>

// out[b,f] = x[b,0,f]*W[f,0] + x[b,1,f]*W[f,1] + bias[f]
// B=8192, DAYS=2, F=4096. Pure HBM-streaming kernel (384 MiB @ 23.3 TB/s ~= 17us).

typedef __attribute__((ext_vector_type(4))) float v4f;

#define BDIM 8192
#define FDIM 4096
#define ROWS 8                 // batch rows per thread (W/bias register reuse)
#define BLOCK 256              // 8 wave32s

__global__ __launch_bounds__(BLOCK) void conv2tap_fused(
    const float* __restrict__ x,     // (B, 2, F)
    const float* __restrict__ W,     // (F, 2) interleaved: w0[f], w1[f]
    const float* __restrict__ bias,  // (F,)
    float* __restrict__ out)         // (B, 1, F)
{
    const int FG = FDIM / 4;                       // 1024 feature groups of float4
    const long id = (long)blockIdx.x * BLOCK + threadIdx.x;
    const int  fg     = (int)(id % FG);            // lanes -> consecutive float4s (coalesced)
    const long bchunk = id / FG;
    const int  f  = fg * 4;
    const long b0 = bchunk * ROWS;

    // W rows f..f+3: 8 consecutive floats (w0,w1 interleaved). Hot, cache-resident -> RT loads.
    v4f wlo = *(const v4f*)(W + 2 * f);            // w0[f],w1[f],w0[f+1],w1[f+1]
    v4f whi = *(const v4f*)(W + 2 * f + 4);        // w0[f+2],w1[f+2],w0[f+3],w1[f+3]
    v4f w0 = { wlo.x, wlo.z, whi.x, whi.z };
    v4f w1 = { wlo.y, wlo.w, whi.y, whi.w };
    v4f bb = *(const v4f*)(bias + f);

    const float* xrow = x   + (size_t)b0 * (2 * FDIM) + f;
    float*       orow = out + (size_t)b0 * FDIM       + f;

#pragma unroll
    for (int r = 0; r < ROWS; ++r) {
        const float* xp = xrow + (size_t)r * (2 * FDIM);

        // Prefetch next batch row's two x lines (emits global_prefetch_b8).
        if (r + 1 < ROWS) {
            __builtin_prefetch(xp + 2 * FDIM,        0, 0);
            __builtin_prefetch(xp + 2 * FDIM + FDIM, 0, 0);
        }

        // Streamed-once data: non-temporal b128 loads / store (TH=NT),
        // keeps the 384 MiB stream out of the way of cached W/bias.
        v4f x0 = __builtin_nontemporal_load((const v4f*)xp);
        v4f x1 = __builtin_nontemporal_load((const v4f*)(xp + FDIM));

        v4f acc = bb;
        acc += x0 * w0;    // v_fmac_f32 x4
        acc += x1 * w1;    // v_fmac_f32 x4

        __builtin_nontemporal_store(acc, (v4f*)(orow + (size_t)r * FDIM));
    }
}

extern "C" void kernel_launch(void* const* d_in, const int* in_sizes, int n_in,
                              void* d_out, int out_size, void* d_ws, size_t ws_size,
                              hipStream_t stream) {
    const float* x    = (const float*)d_in[0];   // (8192, 2, 4096) f32
    const float* W    = (const float*)d_in[1];   // (4096, 2)       f32
    const float* bias = (const float*)d_in[2];   // (4096,)         f32
    float*       out  = (float*)d_out;           // (8192, 1, 4096) f32

    const long total_threads = (long)(FDIM / 4) * (BDIM / ROWS);   // 1,048,576
    const int  blocks = (int)(total_threads / BLOCK);              // 4096

    conv2tap_fused<<<blocks, BLOCK, 0, stream>>>(x, W, bias, out);
}